// GLM_grating_clip_81758997447013
// MI455X (gfx1250) — compile-verified
//
#include <hip/hip_runtime.h>
#include <math.h>

// Problem constants (from reference)
#define NEURONS 54
#define DT      4
#define STEPS   30
#define POOL    8
#define BATCH   16
#define HW_IN   600
#define PH      75              // pooled H = W
#define K_REAL  5625            // 75*75
#define K_PAD   5632            // pad K to multiple of 4
#define KTILES  (K_PAD / 4)     // 1408 WMMA k-steps
#define ROWS    (BATCH*STEPS)   // 480 GEMM rows
#define N_PAD   64              // pad 54 -> 64 (4 WMMA N-tiles)
#define KC      512             // K-chunk staged in LDS
#define LDS_STRIDE 516          // 512+4: bank-conflict-free, rows 16B aligned

typedef __attribute__((ext_vector_type(2))) float v2f;
typedef __attribute__((ext_vector_type(4))) float v4f;
typedef __attribute__((ext_vector_type(8))) float v8f;

// ---------------------------------------------------------------------------
// Kernel 1: 8x8 average pooling, fully streaming (the only HBM-bound part:
// 691 MB -> ~30us floor at 23.3 TB/s). x has zero reuse -> non-temporal loads
// keep L2 clean for fc_w / xp. One thread per padded output element;
// 16 B128 loads per pooled pixel. Zero-pads K so the GEMM A-side is branch-free.
// ---------------------------------------------------------------------------
__global__ void pool_kernel(const float* __restrict__ x, float* __restrict__ xp) {
  size_t gid = (size_t)blockIdx.x * blockDim.x + threadIdx.x;
  const size_t total = (size_t)ROWS * K_PAD;
  if (gid >= total) return;
  const int row = (int)(gid / K_PAD);
  const int k   = (int)(gid % K_PAD);
  if (k >= K_REAL) { xp[gid] = 0.0f; return; }
  const int ph = k / PH, pw = k % PH;
  const float* src = x + (size_t)row * (HW_IN * HW_IN)
                       + (size_t)(ph * POOL) * HW_IN + (size_t)(pw * POOL);
  float s = 0.0f;
#pragma unroll
  for (int r = 0; r < POOL; ++r) {
    const v4f* p = (const v4f*)(src + (size_t)r * HW_IN);
    v4f a = __builtin_nontemporal_load(p);
    v4f b = __builtin_nontemporal_load(p + 1);
    s += a.x + a.y + a.z + a.w + b.x + b.y + b.z + b.w;
  }
  xp[gid] = s * (1.0f / 64.0f);
}

// ---------------------------------------------------------------------------
// Kernel 1b: pre-swizzle fc_w into WMMA B-fragment layout, zero-padded.
// Bp2[(kt*64 + n)*2 + j] = { B[4kt+2j][n], B[4kt+2j+1][n] },  B[k][n]=fc_w[n][k]
// This hoists ALL bounds checks out of the GEMM inner loop (round-1 disasm
// showed exec-mask branches around every B load).
// ---------------------------------------------------------------------------
__global__ void pack_b_kernel(const float* __restrict__ fc_w, float2* __restrict__ Bp2) {
  const int idx = blockIdx.x * blockDim.x + threadIdx.x;   // [0, KTILES*64*2)
  if (idx >= KTILES * N_PAD * 2) return;
  const int kt  = idx >> 7;          // / 128
  const int rem = idx & 127;
  const int n   = rem >> 1;
  const int j   = rem & 1;
  const int k0  = kt * 4 + j * 2;
  float2 v;
  v.x = (n < NEURONS && k0     < K_REAL) ? fc_w[(size_t)n * K_REAL + k0]     : 0.0f;
  v.y = (n < NEURONS && k0 + 1 < K_REAL) ? fc_w[(size_t)n * K_REAL + k0 + 1] : 0.0f;
  Bp2[idx] = v;
}

// ---------------------------------------------------------------------------
// Kernel 2: frames[480][64] = xp[480][K_PAD] @ fc_w^T via V_WMMA_F32_16X16X4_F32.
// grid = 30 M-tiles; 4 waves/WG, one 16-wide N-tile per wave.
// Inner loop per WMMA: ds_load_b64 (A, conflict-free LDS) +
// global_load_b64 (B fragment, L2-resident) + v_wmma. Branch-free.
// Fragment layouts per CDNA5 ISA 7.12.2:
//   A 16x4 f32: v0 = {K=0 | K=2}, v1 = {K=1 | K=3} across lane halves, M = lane%16
//   B 4x16 f32: mirrored (N = lane%16)
//   C/D 16x16 f32: 8 VGPRs, M = v + (lane>=16 ? 8 : 0), N = lane%16
// ---------------------------------------------------------------------------
__global__ void __launch_bounds__(128)
fc_wmma_kernel(const float* __restrict__ xp, const float2* __restrict__ Bp2,
               float* __restrict__ frames) {
  __shared__ float Atile[16 * LDS_STRIDE];
  const int tid  = threadIdx.x;
  const int lane = tid & 31;
  const int wave = tid >> 5;          // 0..3
  const int n0   = wave * 16;
  const int m0   = blockIdx.x * 16;
  const int l16  = lane & 15;
  const bool hi  = lane >= 16;
  const int n    = n0 + l16;

  const float* arow_base = &Atile[l16 * LDS_STRIDE + (hi ? 2 : 0)];

  v8f acc = {};
  for (int kc0 = 0; kc0 < K_PAD; kc0 += KC) {
    // cooperative A-tile load: 16 rows x 512 cols = 2048 float4s over 128 threads
    for (int i = tid; i < 16 * (KC / 4); i += 128) {
      const int r  = i >> 7;          // KC/4 == 128
      const int c4 = i & 127;
      const v4f v = *(const v4f*)(xp + (size_t)(m0 + r) * K_PAD + kc0 + c4 * 4);
      *(v4f*)(&Atile[r * LDS_STRIDE + c4 * 4]) = v;
    }
    __syncthreads();

    const float2* bptr = Bp2 + (((size_t)(kc0 >> 2) * N_PAD + n) * 2 + (hi ? 1 : 0));
#pragma unroll 8
    for (int k = 0; k < KC; k += 4) {
      v2f a;
      a.x = arow_base[k];
      a.y = arow_base[k + 1];
      const float2 bv = *bptr;
      bptr += 2 * N_PAD;              // next k-tile (constant 1KB stride)
      v2f b;
      b.x = bv.x;
      b.y = bv.y;
      acc = __builtin_amdgcn_wmma_f32_16x16x4_f32(
          /*neg_a=*/false, a, /*neg_b=*/false, b,
          /*c_mod=*/(short)0, acc, /*reuse_a=*/false, /*reuse_b=*/false);
    }
    __syncthreads();                  // Atile reuse next chunk
  }
#pragma unroll
  for (int v = 0; v < 8; ++v) {
    const int m = m0 + v + (hi ? 8 : 0);
    frames[(size_t)m * N_PAD + n] = acc[v];
  }
}

// ---------------------------------------------------------------------------
// Kernel 3: sequential 30-step spiking scan. One block, 864 = 16*54 threads
// (27 waves). hist[4] in registers; lateral sum + global mean via LDS.
// ---------------------------------------------------------------------------
__global__ void __launch_bounds__(BATCH * NEURONS)
scan_kernel(const float* __restrict__ frames, const float* __restrict__ fc_b,
            const float* __restrict__ l_weight, const float* __restrict__ h_weight,
            float* __restrict__ out) {
  __shared__ float sh_hsum[BATCH * NEURONS];
  __shared__ float sh_red[1024];
  const int i = threadIdx.x;            // 0..863
  const int b = i / NEURONS;
  const int n = i % NEURONS;
  if (i < 1024 - BATCH * NEURONS) sh_red[BATCH * NEURONS + i] = 0.0f;

  const float lw0 = l_weight[n * DT + 0], lw1 = l_weight[n * DT + 1];
  const float lw2 = l_weight[n * DT + 2], lw3 = l_weight[n * DT + 3];
  const float hw0 = h_weight[0], hw1 = h_weight[1];
  const float hw2 = h_weight[2], hw3 = h_weight[3];
  const float bias = fc_b[n];

  float h0 = 0.f, h1 = 0.f, h2 = 0.f, h3 = 0.f, count = 0.f;
  __syncthreads();

  for (int t = 0; t < STEPS; ++t) {
    const float hs = h0 * lw0 + h1 * lw1 + h2 * lw2 + h3 * lw3;
    sh_hsum[i] = hs;
    __syncthreads();
    float tot = 0.0f;
    for (int j = 0; j < NEURONS; ++j) tot += sh_hsum[b * NEURONS + j];

    const float cur = frames[(size_t)(b * STEPS + t) * N_PAD + n] + bias
                    + (tot - hs) + h0 * hw0 + h1 * hw1 + h2 * hw2 + h3 * hw3;
    const float rate = expf(cur);

    // deterministic tree reduction for the global mean
    sh_red[i] = rate;
    __syncthreads();
    for (int s = 512; s > 0; s >>= 1) {
      if (i < s) sh_red[i] += sh_red[i + s];
      __syncthreads();
    }
    const float thr = sh_red[0] * (1.0f / (BATCH * NEURONS));

    const float spk = (rate > thr) ? 1.0f : 0.0f;
    count += spk;
    h0 = h1; h1 = h2; h2 = h3; h3 = spk;
    __syncthreads();   // protect sh_hsum / sh_red reuse next step
  }
  // softplus, numerically stable for count in [0, 30]
  out[i] = count + log1pf(expf(-count));
}

// ---------------------------------------------------------------------------
extern "C" void kernel_launch(void* const* d_in, const int* in_sizes, int n_in,
                              void* d_out, int out_size, void* d_ws, size_t ws_size,
                              hipStream_t stream) {
  const float* x   = (const float*)d_in[0];   // [16,30,600,600]
  const float* fcw = (const float*)d_in[1];   // [54,5625]
  const float* fcb = (const float*)d_in[2];   // [54]
  const float* lw  = (const float*)d_in[3];   // [54,4]
  const float* hw  = (const float*)d_in[4];   // [4]
  float* out = (float*)d_out;                 // [16,54]

  float*  xp     = (float*)d_ws;                          // 480*5632 f32 (~10.3 MB)
  float*  frames = xp + (size_t)ROWS * K_PAD;             // 480*64 f32
  float2* Bp2    = (float2*)(frames + (size_t)ROWS * N_PAD); // 1408*64*2 float2 (~1.4 MB)

  const size_t total = (size_t)ROWS * K_PAD;
  pool_kernel<<<(unsigned)((total + 255) / 256), 256, 0, stream>>>(x, xp);
  const int bElems = KTILES * N_PAD * 2;
  pack_b_kernel<<<(bElems + 255) / 256, 256, 0, stream>>>(fcw, Bp2);
  fc_wmma_kernel<<<ROWS / 16, 128, 0, stream>>>(xp, Bp2, frames);
  scan_kernel<<<1, BATCH * NEURONS, 0, stream>>>(frames, fcb, lw, hw, out);
}